// NonLocalInteraction_37417755082832
// MI455X (gfx1250) — compile-verified
//
#include <hip/hip_runtime.h>
#include <hip/hip_bf16.h>
#include <cstddef>

typedef __attribute__((ext_vector_type(2))) float v2f;
typedef __attribute__((ext_vector_type(8))) float v8f;
typedef int v4i_vec __attribute__((vector_size(16)));   // 4 x i32, b128 payload

#define DIM 256
#define NROWS 30720
#define PROJ_STRIDE (NROWS * DIM)      // 7,864,320 floats per Q/K/V matrix
#define WT_FLOATS (9 * DIM * DIM)      // 589,824 transposed-weight floats
#define QKV_OFF (1u << 20)             // scratch offset (floats) past WT region

__device__ __forceinline__ v8f wmma_f32x4(v2f a, v2f b, v8f c) {
  // D = A(16x4 f32) x B(4x16 f32) + C(16x16 f32)
  return __builtin_amdgcn_wmma_f32_16x16x4_f32(false, a, false, b, (short)0, c,
                                               false, false);
}

__device__ __forceinline__ float swishf(float x, float al, float gm) {
  // alpha * x * sigmoid(gamma * x), fast rcp (v_rcp_f32) + fast exp
  return al * x * __builtin_amdgcn_rcpf(1.0f + __expf(-gm * x));
}

// ---- CDNA5 async global->LDS copy (ASYNCcnt) with sync fallback ------------
__device__ __forceinline__ void async_b128(float* ldst, const float* gsrc) {
#if __has_builtin(__builtin_amdgcn_global_load_async_to_lds_b128)
  __builtin_amdgcn_global_load_async_to_lds_b128(
      (__attribute__((address_space(1))) v4i_vec*)gsrc,
      (__attribute__((address_space(3))) v4i_vec*)ldst, 0, 0);
#else
  *(float4*)ldst = *(const float4*)gsrc;
#endif
}
__device__ __forceinline__ void wait_async_0() {
#if __has_builtin(__builtin_amdgcn_s_wait_asynccnt)
  __builtin_amdgcn_s_wait_asynccnt(0);
#else
  asm volatile("s_wait_asynccnt 0x0" ::: "memory");
#endif
}
__device__ __forceinline__ void wait_async_16() {
#if __has_builtin(__builtin_amdgcn_s_wait_asynccnt)
  __builtin_amdgcn_s_wait_asynccnt(16);
#else
  asm volatile("s_wait_asynccnt 0x10" ::: "memory");
#endif
}

// ---------- kernel 1: weight transpose into pairwise layout -----------------
// wt2[mat][(k>>1)*512 + e*2 + (k&1)] = W[mat%3][e][k]
// so a WMMA B-fragment {k, k+1} x e is one contiguous float2.
__global__ __launch_bounds__(256) void wt_transpose_kernel(
    const float* __restrict__ Wq, const float* __restrict__ Wk,
    const float* __restrict__ Wv, float* __restrict__ wt) {
  int gid = blockIdx.x * 256 + threadIdx.x;
  if (gid >= WT_FLOATS) return;
  int mat = gid >> 16;              // 0..8
  int rem = gid & 0xFFFF;
  int k = rem >> 8;
  int e = rem & 0xFF;
  const float* W = (mat < 3) ? Wq : (mat < 6) ? Wk : Wv;
  int i = mat % 3;
  wt[mat * 65536 + (k >> 1) * 512 + e * 2 + (k & 1)] =
      W[i * DIM * DIM + e * DIM + k];
}

// ---------- kernel 2: fused resMLP for Q, K, V ------------------------------
// grid: (NROWS/64, 3), block: 128 threads = 4 waves, 16 rows per wave.
// Double-buffered async weight-chunk staging in LDS shared by the 4 waves.
__global__ __launch_bounds__(128) void qkv_resmlp_kernel(
    const float* __restrict__ x, const float* __restrict__ wt,
    const float* __restrict__ bq, const float* __restrict__ aq,
    const float* __restrict__ gq, const float* __restrict__ bk,
    const float* __restrict__ ak, const float* __restrict__ gk,
    const float* __restrict__ bv, const float* __restrict__ av,
    const float* __restrict__ gv, float* __restrict__ qkv) {
  __shared__ float sX[4][16 * DIM];     // raw input rows (residual source)
  __shared__ float sT[4][16 * DIM];     // chained intermediate (t1 then h)
  __shared__ float sW[2][32 * DIM];     // double-buffered weight k-chunks

  const int proj = blockIdx.y;
  const float* bias = (proj == 0) ? bq : (proj == 1) ? bk : bv;
  const float* alp  = (proj == 0) ? aq : (proj == 1) ? ak : av;
  const float* gam  = (proj == 0) ? gq : (proj == 1) ? gk : gv;
  const float* wtp = wt + (size_t)proj * 3 * DIM * DIM;
  float* outP = qkv + (size_t)proj * PROJ_STRIDE;

  const int tid = threadIdx.x;
  const int wave = tid >> 5;
  const int lane = tid & 31;
  const int half = lane >> 4;
  const int lr = lane & 15;
  const int rowBase = blockIdx.x * 64 + wave * 16;

  // async-stage this wave's 16 input rows into LDS
  {
    const float* src = x + (size_t)rowBase * DIM;
#pragma unroll
    for (int i = 0; i < 32; ++i)
      async_b128(&sX[wave][(lane + i * 32) * 4], src + (lane + i * 32) * 4);
  }

  float aSv[3], gSv[3];
#pragma unroll
  for (int g = 0; g < 3; ++g) { aSv[g] = alp[g]; gSv[g] = gam[g]; }

#pragma unroll  // full unroll: g is compile-time -> branch-free epilogues
  for (int g = 0; g < 3; ++g) {
    const float aS = aSv[g], gS = gSv[g];
    const float* wg = wtp + g * DIM * DIM;
    const float* src = (g == 0) ? sX[wave] : sT[wave];
    v8f acc[16];
#pragma unroll
    for (int et = 0; et < 16; ++et)
#pragma unroll
      for (int v = 0; v < 8; ++v) acc[et][v] = 0.0f;

    // issue chunk 0 -> buffer 0
#pragma unroll
    for (int i = 0; i < 16; ++i)
      async_b128(&sW[0][(tid + i * 128) * 4], wg + (tid + i * 128) * 4);

#pragma unroll 1
    for (int kc = 0; kc < 8; ++kc) {            // 8 chunks of 32 k-values
      const int cur = kc & 1;
      if (kc < 7) {                             // prefetch next chunk
        const float* gsrc = wg + (kc + 1) * 32 * DIM;
#pragma unroll
        for (int i = 0; i < 16; ++i)
          async_b128(&sW[1 - cur][(tid + i * 128) * 4],
                     gsrc + (tid + i * 128) * 4);
        wait_async_16();                        // chunk kc (and older) landed
      } else {
        wait_async_0();
      }
      __syncthreads();
#pragma unroll 1
      for (int ks = 0; ks < 8; ++ks) {
        const int k0 = ks * 4;
        // A fragment: swish(activation), b64 LDS read, reused over 16 e-tiles
        const float2 af =
            *(const float2*)(src + lr * DIM + kc * 32 + k0 + 2 * half);
        v2f a;
        a.x = swishf(af.x, aS, gS);
        a.y = swishf(af.y, aS, gS);
        const float* bp = sW[cur] + (2 * ks + half) * 512 + lr * 2;
#pragma unroll
        for (int et = 0; et < 16; ++et) {       // pairwise layout: 1 ds_b64
          const float2 bf = *(const float2*)(bp + et * 32);
          v2f b;
          b.x = bf.x;
          b.y = bf.y;
          acc[et] = wmma_f32x4(a, b, acc[et]);
        }
      }
      __syncthreads();   // all waves done with sW[cur] before it is re-filled
    }

    if (g < 2) {  // write t1 (g=0) / h = x + t2 (g=1) back to LDS
#pragma unroll
      for (int et = 0; et < 16; ++et) {
        const float bb = bias[g * DIM + et * 16 + lr];
#pragma unroll
        for (int v = 0; v < 8; ++v) {
          float t = acc[et][v] + bb;
          if (g == 1) t += sX[wave][(v + 8 * half) * DIM + et * 16 + lr];
          sT[wave][(v + 8 * half) * DIM + et * 16 + lr] = t;
        }
      }
    } else {      // final projection -> scratch
#pragma unroll
      for (int et = 0; et < 16; ++et) {
        const float bb = bias[2 * DIM + et * 16 + lr];
#pragma unroll
        for (int v = 0; v < 8; ++v)
          outP[(size_t)(rowBase + v + 8 * half) * DIM + et * 16 + lr] =
              acc[et][v] + bb;
      }
    }
  }
}

// ---------- kernel 3: ragged flash attention --------------------------------
// lengths = tile([256,512,768,1024],12); blocks of 4 query tiles (64 queries).
__global__ __launch_bounds__(128) void attn_kernel(const float* __restrict__ qkv,
                                                   float* __restrict__ out) {
  __shared__ float sKtp[128 * 32];    // K tile, transposed pairwise:
                                      // [(d>>1)][kk*2 + (d&1)]
  __shared__ float sV[16 * DIM];      // V tile: V[kk][e]
  __shared__ float sP[4][16 * 16];    // per-wave prob tile relayout buffer

  const float* Q = qkv;
  const float* K = qkv + (size_t)PROJ_STRIDE;
  const float* V = qkv + (size_t)2 * PROJ_STRIDE;

  const int bid = blockIdx.x;
  const int group = bid / 40;         // 40 blocks per 4-molecule group
  const int r = bid - group * 40;
  int m, cpre;
  if (r < 4)       { m = 0; cpre = 0;  }
  else if (r < 12) { m = 1; cpre = 4;  }
  else if (r < 24) { m = 2; cpre = 12; }
  else             { m = 3; cpre = 24; }
  const int L = 256 * (m + 1);
  const int rowpre = (m == 0) ? 0 : (m == 1) ? 256 : (m == 2) ? 768 : 1536;
  const int base = group * 2560 + rowpre;   // molecule's flat row offset
  const int qchunk = r - cpre;

  const int tid = threadIdx.x;
  const int wave = tid >> 5;
  const int lane = tid & 31;
  const int half = lane >> 4;
  const int lr = lane & 15;
  const int qrow = base + qchunk * 64 + wave * 16;

  // Q tile pre-loaded in WMMA A-layout, pre-scaled by 1/sqrt(256)
  v2f qa[64];
  {
    const float* qp = Q + (size_t)(qrow + lr) * DIM + 2 * half;
#pragma unroll
    for (int j = 0; j < 64; ++j) {
      float2 t = *(const float2*)(qp + 4 * j);
      qa[j].x = t.x * 0.0625f;
      qa[j].y = t.y * 0.0625f;
    }
  }

  v8f O[16];
#pragma unroll
  for (int et = 0; et < 16; ++et)
#pragma unroll
    for (int v = 0; v < 8; ++v) O[et][v] = 0.0f;
  float mrow[8], lrow[8];
#pragma unroll
  for (int v = 0; v < 8; ++v) { mrow[v] = -3.0e38f; lrow[v] = 0.0f; }

  const int nkt = L >> 4;
#pragma unroll 1
  for (int kt = 0; kt < nkt; ++kt) {
    __syncthreads();   // previous iteration's sKtp/sV reads complete
    {
      const float4* k4 = (const float4*)(K + (size_t)(base + kt * 16) * DIM);
      const float* vg = V + (size_t)(base + kt * 16) * DIM;
      // V tile: straight async copy global->LDS
#pragma unroll
      for (int i = 0; i < 8; ++i) {
        const int j = tid + i * 128;
        async_b128(&sV[j * 4], vg + j * 4);
      }
      // K tile: transpose into pairwise layout (two b64 LDS stores per b128)
#pragma unroll
      for (int i = 0; i < 8; ++i) {
        const int j = tid + i * 128;
        const float4 kk = k4[j];
        const int row = j >> 6;
        const int col = (j & 63) * 4;
        float2 lo, hi;
        lo.x = kk.x; lo.y = kk.y;
        hi.x = kk.z; hi.y = kk.w;
        *(float2*)&sKtp[(col >> 1) * 32 + row * 2] = lo;
        *(float2*)&sKtp[((col >> 1) + 1) * 32 + row * 2] = hi;
      }
    }
    wait_async_0();
    __syncthreads();

    // S = (Q/16) . K^T   (64 f32 WMMA k-steps over d, 1 ds_b64 per B frag)
    v8f s;
#pragma unroll
    for (int v = 0; v < 8; ++v) s[v] = 0.0f;
#pragma unroll
    for (int j = 0; j < 64; ++j) {
      const float2 bf = *(const float2*)(sKtp + (2 * j + half) * 32 + lr * 2);
      v2f b;
      b.x = bf.x;
      b.y = bf.y;
      s = wmma_f32x4(qa[j], b, s);
    }

    // online softmax: per-row (m = v + 8*half) reduction across 16 lanes
    float scale[8];
#pragma unroll
    for (int v = 0; v < 8; ++v) {
      float rm = s[v];
      rm = fmaxf(rm, __shfl_xor(rm, 1, 16));
      rm = fmaxf(rm, __shfl_xor(rm, 2, 16));
      rm = fmaxf(rm, __shfl_xor(rm, 4, 16));
      rm = fmaxf(rm, __shfl_xor(rm, 8, 16));
      const float mn = fmaxf(mrow[v], rm);
      scale[v] = __expf(mrow[v] - mn);
      mrow[v] = mn;
      const float p = __expf(s[v] - mn);
      s[v] = p;
      float rs = p;
      rs += __shfl_xor(rs, 1, 16);
      rs += __shfl_xor(rs, 2, 16);
      rs += __shfl_xor(rs, 4, 16);
      rs += __shfl_xor(rs, 8, 16);
      lrow[v] = lrow[v] * scale[v] + rs;
    }
#pragma unroll
    for (int et = 0; et < 16; ++et)
#pragma unroll
      for (int v = 0; v < 8; ++v) O[et][v] *= scale[v];

    // relayout P (C-layout -> A-layout) through per-wave LDS tile
#pragma unroll
    for (int v = 0; v < 8; ++v) sP[wave][(v + 8 * half) * 16 + lr] = s[v];

    // O += P . V  (4 f32 WMMA k-steps over the 16 keys)
#pragma unroll
    for (int ks = 0; ks < 4; ++ks) {
      const int k0 = ks * 4;
      const float2 pf = *(const float2*)(sP[wave] + lr * 16 + k0 + 2 * half);
      v2f a;
      a.x = pf.x;
      a.y = pf.y;
      const float* vp = sV + (k0 + 2 * half) * DIM;
#pragma unroll
      for (int et = 0; et < 16; ++et) {
        v2f b;
        b.x = vp[et * 16 + lr];
        b.y = vp[DIM + et * 16 + lr];
        O[et] = wmma_f32x4(a, b, O[et]);
      }
    }
  }

  float inv[8];
#pragma unroll
  for (int v = 0; v < 8; ++v) inv[v] = __builtin_amdgcn_rcpf(lrow[v]);
#pragma unroll
  for (int et = 0; et < 16; ++et)
#pragma unroll
    for (int v = 0; v < 8; ++v)
      out[(size_t)(qrow + v + 8 * half) * DIM + et * 16 + lr] =
          O[et][v] * inv[v];
}

// ---------------------------------------------------------------------------
extern "C" void kernel_launch(void* const* d_in, const int* in_sizes, int n_in,
                              void* d_out, int out_size, void* d_ws,
                              size_t ws_size, hipStream_t stream) {
  const float* x  = (const float*)d_in[0];
  const float* Wq = (const float*)d_in[5];
  const float* bq = (const float*)d_in[6];
  const float* aq = (const float*)d_in[7];
  const float* gq = (const float*)d_in[8];
  const float* Wk = (const float*)d_in[9];
  const float* bk = (const float*)d_in[10];
  const float* ak = (const float*)d_in[11];
  const float* gk = (const float*)d_in[12];
  const float* Wv = (const float*)d_in[13];
  const float* bv = (const float*)d_in[14];
  const float* av = (const float*)d_in[15];
  const float* gv = (const float*)d_in[16];

  float* ws = (float*)d_ws;
  float* wt = ws;                 // pairwise transposed weights [9][256][256]
  float* qkv = ws + QKV_OFF;      // Q, K, V each PROJ_STRIDE floats

  wt_transpose_kernel<<<(WT_FLOATS + 255) / 256, 256, 0, stream>>>(Wq, Wk, Wv,
                                                                   wt);
  qkv_resmlp_kernel<<<dim3(NROWS / 64, 3), 128, 0, stream>>>(
      x, wt, bq, aq, gq, bk, ak, gk, bv, av, gv, qkv);
  attn_kernel<<<480, 128, 0, stream>>>(qkv, (float*)d_out);
}